// PrettyPCF_5068061409534
// MI455X (gfx1250) — compile-verified
//
#include <hip/hip_runtime.h>
#include <math.h>

#define NPOINTS 1536
#define NBBINS  50
#define NTILE   96                     // NPOINTS / 16
#define TWO_PI_F 6.28318530717958647f
#define PI_F     3.14159265358979323f
#define GF_F     2.25675833419102515f  // 1/(sqrt(pi)*0.25)

typedef float v2f __attribute__((ext_vector_type(2)));
typedef float v8f __attribute__((ext_vector_type(8)));

__device__ __forceinline__ float rmax_f() {
    // RMAX = 2*sqrt(1/(2*sqrt(3)*NPOINTS))
    return 2.0f * sqrtf(1.0f / (2.0f * sqrtf(3.0f) * (float)NPOINTS));
}

// ---------------------------------------------------------------------------
// Kernel 0: zero the 50-bin global accumulator (d_ws scratch is poisoned).
// ---------------------------------------------------------------------------
__global__ void pcf_init_bins(float* bins) {
    int t = threadIdx.x;
    if (t < NBBINS) bins[t] = 0.0f;
}

// ---------------------------------------------------------------------------
// Kernel 1: pairwise distances via V_WMMA_F32_16X16X4_F32.
// One wave per 16x16 tile of the 1536x1536 distance matrix.
//   dot = A(16x4) x B(4x16), K=2 real (x,y), K=2..3 zero-padded.
//   d   = sqrt(max(0, |a|^2 + |b|^2 - 2*dot))
// A-frag (32-bit 16x4): lanes 0-15 hold K=0,1 ; lanes 16-31 hold K=2,3 (=0).
// B-frag (4x16):        lanes 0-15 hold rows K=0,1 ; lanes 16-31 rows 2,3 (=0).
// D (16x16 f32): lane l, vgpr v -> row M = v + 8*(l>=16), col N = l%16.
// Stored column-major dmat[j*N + i] so the accumulation pass reads coalesced.
// ---------------------------------------------------------------------------
__global__ void pcf_dist_wmma(const float* __restrict__ A,
                              const float* __restrict__ B,
                              float* __restrict__ dmat) {
    const int wave = threadIdx.x >> 5;
    const int lane = threadIdx.x & 31;
    const int tile = blockIdx.x * (blockDim.x >> 5) + wave; // 0 .. NTILE*NTILE-1
    const int ti = tile / NTILE;
    const int tj = tile - ti * NTILE;
    const int ibase = ti << 4;
    const int jbase = tj << 4;
    const int half = lane >> 4;     // 0: real K rows, 1: zero-pad K rows
    const int l16  = lane & 15;

    v2f af, bf;
    if (half == 0) {
        const int ia = (ibase + l16) * 3;
        af.x = A[ia + 0]; af.y = A[ia + 1];
        const int ib = (jbase + l16) * 3;
        bf.x = B[ib + 0]; bf.y = B[ib + 1];
    } else {
        af.x = 0.0f; af.y = 0.0f;
        bf.x = 0.0f; bf.y = 0.0f;
    }

    v8f c = {};
    // D = A x B + 0  -> emits v_wmma_f32_16x16x4_f32
    v8f dot = __builtin_amdgcn_wmma_f32_16x16x4_f32(
        /*neg_a=*/false, af, /*neg_b=*/false, bf,
        /*c_mod=*/(short)0, c, /*reuse_a=*/false, /*reuse_b=*/false);

    // |b_j|^2 for this lane's column
    const int jb = (jbase + l16) * 3;
    const float bx = B[jb + 0], by = B[jb + 1];
    const float nb = bx * bx + by * by;

#pragma unroll
    for (int v = 0; v < 8; ++v) {
        const int i = ibase + v + 8 * half;
        const int ia = i * 3;
        const float ax = A[ia + 0], ay = A[ia + 1];
        const float na = ax * ax + ay * ay;
        const float d2 = na + nb - 2.0f * dot[v];
        dmat[(size_t)(jbase + l16) * NPOINTS + i] = sqrtf(fmaxf(d2, 0.0f));
    }
}

// ---------------------------------------------------------------------------
// Kernel 2: per A-point Gaussian binning + perimeter weight + reduction.
// One thread per A-point; 50-bin accumulator lives in registers; inner loop
// is pure VALU (v_exp_f32). Reduction via global_atomic_add_f32.
// ---------------------------------------------------------------------------
__global__ void pcf_accum(const float* __restrict__ A,
                          const float* __restrict__ dmat,
                          const int* __restrict__ scPtr,
                          float* __restrict__ bins) {
    const int i = blockIdx.x * blockDim.x + threadIdx.x;
    if (i >= NPOINTS) return;

    const int same = *scPtr;

    const float RMAX = rmax_f();
    const float binw = 0.1f * RMAX;             // (N_RMAX/NBBINS)*RMAX
    const float cexp = 16.0f / (RMAX * RMAX);   // 1/(sigma^2 * RMAX^2), sigma=0.25

    float acc[NBBINS];
#pragma unroll
    for (int b = 0; b < NBBINS; ++b) acc[b] = 0.0f;

    for (int j = 0; j < NPOINTS; ++j) {
        const float d = dmat[(size_t)j * NPOINTS + i];
        if (same && (j == i)) continue;          // (1 - eye) diagonal mask
#pragma unroll
        for (int b = 0; b < NBBINS; ++b) {
            const float z = binw * (float)(b + 1) - d;
            acc[b] += __expf(-z * z * cexp);
        }
    }

    // Perimeter weight for point i (x,y in [0,1]^2), clipped to [0,4].
    const float x = A[i * 3 + 0];
    const float y = A[i * 3 + 1];
    float dxs[4], dys[4], a1s[4], a2s[4];
    dxs[0] = x;        dys[0] = y;
    dxs[1] = 1.0f - x; dys[1] = y;
    dxs[2] = y;        dys[2] = x;
    dxs[3] = 1.0f - y; dys[3] = x;
#pragma unroll
    for (int s = 0; s < 4; ++s) {
        a1s[s] = atan2f(dys[s], dxs[s]);
        a2s[s] = atan2f(1.0f - dys[s], dxs[s]);
    }

#pragma unroll 2
    for (int b = 0; b < NBBINS; ++b) {
        const float rs = binw * (float)(b + 1);
        float full = TWO_PI_F;
#pragma unroll
        for (int s = 0; s < 4; ++s) {
            if (rs > dxs[s]) {
                float ratio = dxs[s] / rs;
                ratio = fminf(fmaxf(ratio, -1.0f), 1.0f);
                const float alpha = acosf(ratio);
                full -= fminf(alpha, a1s[s]) + fminf(alpha, a2s[s]);
            }
        }
        float per = fminf(fmaxf(full / TWO_PI_F, 0.0f), 1.0f);
        float w = 1.0f / fmaxf(per, 1e-9f);
        w = fminf(fmaxf(w, 0.0f), 4.0f);
        atomicAdd(&bins[b], acc[b] * GF_F * w);  // global_atomic_add_f32
    }
}

// ---------------------------------------------------------------------------
// Kernel 3: finalize: out[b] = { rs/RMAX , sum / na / (AREA*nb) }
// ---------------------------------------------------------------------------
__global__ void pcf_finalize(const float* __restrict__ bins,
                             float* __restrict__ out) {
    const int b = threadIdx.x;
    if (b < NBBINS) {
        const float RMAX = rmax_f();
        const float rs = 0.1f * RMAX * (float)(b + 1);
        const float inner = fmaxf(0.0f, rs - 0.5f * RMAX);
        const float outer = rs + 0.5f * RMAX;
        const float area = PI_F * (outer * outer - inner * inner);
        out[2 * b + 0] = 0.1f * (float)(b + 1);  // rs / RMAX
        out[2 * b + 1] = bins[b] / (float)NPOINTS / (area * (float)NPOINTS);
    }
}

// ---------------------------------------------------------------------------
extern "C" void kernel_launch(void* const* d_in, const int* in_sizes, int n_in,
                              void* d_out, int out_size, void* d_ws, size_t ws_size,
                              hipStream_t stream) {
    (void)in_sizes; (void)n_in; (void)out_size; (void)ws_size;
    const float* A  = (const float*)d_in[0];   // disks_a [1536,3]
    const float* B  = (const float*)d_in[1];   // disks_b [1536,3]
    const int*   sc = (const int*)d_in[2];     // same_category scalar
    float* out = (float*)d_out;                // [50,2] f32

    float* dmat = (float*)d_ws;                             // 1536*1536 f32 (9.4 MB)
    float* bins = dmat + (size_t)NPOINTS * NPOINTS;         // 50 f32

    pcf_init_bins<<<1, 64, 0, stream>>>(bins);

    // 96*96 = 9216 tiles, 8 waves (256 threads) per block -> 1152 blocks
    pcf_dist_wmma<<<(NTILE * NTILE) / 8, 256, 0, stream>>>(A, B, dmat);

    pcf_accum<<<NPOINTS / 256, 256, 0, stream>>>(A, dmat, sc, bins);

    pcf_finalize<<<1, 64, 0, stream>>>(bins, out);
}